// Attention_layer_12249246728743
// MI455X (gfx1250) — compile-verified
//
#include <hip/hip_runtime.h>

typedef __attribute__((ext_vector_type(2))) float v2f;
typedef __attribute__((ext_vector_type(8))) float v8f;
typedef int vs4i __attribute__((vector_size(16)));   // matches builtin's b128 operand type

#define C_DIM 512
#define P_DIM 3136   // 56*56
#define HW 56
#define KK 49
#define CH 32                 // K-chunk rows staged in LDS
#define NCH (C_DIM / CH)      // 16 chunks
#define LDST 72               // LDS row stride in floats (64 + 8 pad: halves hit disjoint banks)

#define AS1 __attribute__((address_space(1)))
#define AS3 __attribute__((address_space(3)))

#if defined(__has_builtin)
#  if __has_builtin(__builtin_amdgcn_global_load_async_to_lds_b128)
#    define USE_ASYNC_LDS 1
#  endif
#endif
#ifndef USE_ASYNC_LDS
#  define USE_ASYNC_LDS 0
#endif

__device__ __forceinline__ void wait_async0() {
#if USE_ASYNC_LDS
#  if __has_builtin(__builtin_amdgcn_s_wait_asynccnt)
    __builtin_amdgcn_s_wait_asynccnt(0);
#  else
    asm volatile("s_wait_asynccnt 0x0" ::: "memory");
#  endif
#endif
}

__device__ __forceinline__ float wave_sum(float v) {
#pragma unroll
    for (int m = 16; m >= 1; m >>= 1) v += __shfl_xor(v, m, 32);
    return v;
}

// ---------------------------------------------------------------------------
// Kernel 1: Q/K/V 1x1-conv GEMMs via V_WMMA_F32_16X16X4_F32.
// Block = 8 waves -> 128(O) x 64(P) tile; the X (B-operand) 32x64 K-chunk is
// staged once in LDS (async DMA when available) and shared by all 8 waves,
// double-buffered. Each wave: 16x64 tile, 4 accumulators, A reused 4x.
// ---------------------------------------------------------------------------
__global__ __launch_bounds__(256) void qkv_gemm_wmma(
        const float* __restrict__ x,
        const float* __restrict__ qw, const float* __restrict__ qb,
        const float* __restrict__ kw, const float* __restrict__ kb,
        const float* __restrict__ vw, const float* __restrict__ vb,
        float* __restrict__ ws) {
    __shared__ float sx[2][CH][LDST];

    int bid = blockIdx.x;                 // 3 * 4 * 49 = 588 blocks
    int mat = bid / 196;
    int r   = bid % 196;
    int ob  = r / 49;                     // o-block: 128 rows
    int pt  = r % 49;                     // p-tile: 64 cols
    int warp = threadIdx.x >> 5;
    int lane = threadIdx.x & 31;
    int o0 = ob * 128 + warp * 16;
    int p0 = pt * 64;
    int tid = threadIdx.x;

    const float* W  = (mat == 0) ? qw : (mat == 1) ? kw : vw;
    const float* Bv = (mat == 0) ? qb : (mat == 1) ? kb : vb;
    float* out = ws + (size_t)mat * C_DIM * P_DIM;

    // A fragment (16x4 f32): lanes 0-15 -> K={0,1}, lanes 16-31 -> K={2,3}; M = lane&15
    // B fragment (4x16 f32): vgpr r -> K = r + 2*(lane>>4); N = lane&15
    int khalf = lane >> 4;
    int n     = lane & 15;
    const float* wrow = W + (size_t)(o0 + (lane & 15)) * C_DIM + khalf * 2;

    // Stage one 32x64 chunk of X into LDS buffer (512 float4s; 2 per thread).
    auto stage = [&](int ch) {
        int buf = ch & 1;
        const float* src = x + (size_t)(ch * CH) * P_DIM + p0;
#pragma unroll
        for (int e = 0; e < 2; ++e) {
            int id  = tid + e * 256;          // 0..511
            int row = id >> 4;                // 0..31
            int c4  = (id & 15) << 2;         // 0,4,...,60
#if USE_ASYNC_LDS
            __builtin_amdgcn_global_load_async_to_lds_b128(
                (AS1 vs4i*)(src + (size_t)row * P_DIM + c4),
                (AS3 vs4i*)&sx[buf][row][c4], 0, 0);
#else
            float4 v = *(const float4*)(src + (size_t)row * P_DIM + c4);
            *(float4*)&sx[buf][row][c4] = v;
#endif
        }
    };

    v8f acc0 = {0.f,0.f,0.f,0.f,0.f,0.f,0.f,0.f};
    v8f acc1 = acc0, acc2 = acc0, acc3 = acc0;

    stage(0);
    for (int ch = 0; ch < NCH; ++ch) {
        int buf = ch & 1;
        wait_async0();                        // my stage(ch) stores landed
        __syncthreads();                      // everyone staged buf, everyone done reading other buf
        if (ch + 1 < NCH) stage(ch + 1);      // fill other buffer while computing
#pragma unroll
        for (int cc = 0; cc < CH; cc += 4) {
            v2f a = *(const v2f*)(wrow + ch * CH + cc);
            const float* b_lo = &sx[buf][cc + 2 * khalf][n];
            const float* b_hi = b_lo + LDST;
            v2f b0, b1, b2, b3;
            b0.x = b_lo[0];  b0.y = b_hi[0];
            b1.x = b_lo[16]; b1.y = b_hi[16];
            b2.x = b_lo[32]; b2.y = b_hi[32];
            b3.x = b_lo[48]; b3.y = b_hi[48];
            acc0 = __builtin_amdgcn_wmma_f32_16x16x4_f32(false, a, false, b0, (short)0, acc0, false, false);
            acc1 = __builtin_amdgcn_wmma_f32_16x16x4_f32(false, a, false, b1, (short)0, acc1, false, false);
            acc2 = __builtin_amdgcn_wmma_f32_16x16x4_f32(false, a, false, b2, (short)0, acc2, false, false);
            acc3 = __builtin_amdgcn_wmma_f32_16x16x4_f32(false, a, false, b3, (short)0, acc3, false, false);
        }
    }

    // C/D layout: vgpr r -> M = r + 8*(lane>>4); N = lane&15
#pragma unroll
    for (int rr = 0; rr < 8; ++rr) {
        int mr = o0 + rr + 8 * khalf;
        float bb = Bv[mr];
        float* orow = out + (size_t)mr * P_DIM + p0 + n;
        orow[0]  = acc0[rr] + bb;
        orow[16] = acc1[rr] + bb;
        orow[32] = acc2[rr] + bb;
        orow[48] = acc3[rr] + bb;
    }
}

// ---------------------------------------------------------------------------
// Kernel 2: positional bias table bias[i,j] = sum_c h_pos[c,i] + sum_c w_pos[c,j]
// ---------------------------------------------------------------------------
__global__ void pos_bias_kernel(const float* __restrict__ hp,
                                const float* __restrict__ wp,
                                float* __restrict__ bias49) {
    int t = threadIdx.x;
    if (t >= KK) return;
    int i = t / 7, j = t % 7;
    float s = 0.f;
    for (int c = 0; c < C_DIM / 2; ++c) s += hp[c * 7 + i] + wp[c * 7 + j];
    bias49[t] = s;
}

// ---------------------------------------------------------------------------
// Kernel 3: scrambled 7x7 local attention (torch reshape-interleave semantics).
// Group g=(h,x',y') maps to a single channel c* and 64 contiguous positions.
// One wave per group; lane handles d=lane and d=lane+32. Out-of-image K/V are
// the conv biases k_b[c], v_b[c] (conv of zero padding).
// ---------------------------------------------------------------------------
__global__ __launch_bounds__(256) void local_attn_kernel(
        const float* __restrict__ ws,
        const float* __restrict__ kbias, const float* __restrict__ vbias,
        float* __restrict__ out) {
    int wid  = blockIdx.x * (blockDim.x >> 5) + (threadIdx.x >> 5);
    int lane = threadIdx.x & 31;
    if (wid >= 8 * P_DIM) return;             // exact grid; never taken

    int h   = wid / P_DIM;
    int rem = wid % P_DIM;
    int xg  = rem / HW;
    int yg  = rem % HW;
    int B0  = xg * 448 + yg * 64;
    int c   = h * 64 + xg + B0 / P_DIM;
    int p0  = B0 % P_DIM;

    const float* Q  = ws;
    const float* Kc = ws + (size_t)C_DIM * P_DIM;
    const float* Vc = ws + (size_t)2 * C_DIM * P_DIM;
    const float* bias49 = ws + (size_t)3 * C_DIM * P_DIM;

    size_t cb = (size_t)c * P_DIM;
    int pA = p0 + lane, pB = p0 + lane + 32;
    float qA = Q[cb + pA], qB = Q[cb + pB];
    float qsum = wave_sum(qA + qB);

    int axA = pA / HW, ayA = pA % HW;
    int axB = pB / HW, ayB = pB % HW;
    float kbc = kbias[c], vbc = vbias[c];

    float sc[KK];
#pragma unroll
    for (int i = 0; i < 7; ++i) {
#pragma unroll
        for (int j = 0; j < 7; ++j) {
            int nxA = axA + i - 3, nyA = ayA + j - 3;
            int nxB = axB + i - 3, nyB = ayB + j - 3;
            float kA = ((unsigned)nxA < HW && (unsigned)nyA < HW) ? Kc[cb + nxA * HW + nyA] : kbc;
            float kB = ((unsigned)nxB < HW && (unsigned)nyB < HW) ? Kc[cb + nxB * HW + nyB] : kbc;
            float s = wave_sum(qA * kA + qB * kB);
            sc[i * 7 + j] = s + qsum * bias49[i * 7 + j];
        }
    }

    float mx = sc[0];
#pragma unroll
    for (int t = 1; t < KK; ++t) mx = fmaxf(mx, sc[t]);
    float Z = 0.f;
#pragma unroll
    for (int t = 0; t < KK; ++t) { sc[t] = __expf(sc[t] - mx); Z += sc[t]; }
    float invZ = 1.0f / Z;

    float aA = 0.f, aB = 0.f;
#pragma unroll
    for (int i = 0; i < 7; ++i) {
#pragma unroll
        for (int j = 0; j < 7; ++j) {
            float w = sc[i * 7 + j];
            int nxA = axA + i - 3, nyA = ayA + j - 3;
            int nxB = axB + i - 3, nyB = ayB + j - 3;
            float vA = ((unsigned)nxA < HW && (unsigned)nyA < HW) ? Vc[cb + nxA * HW + nyA] : vbc;
            float vB = ((unsigned)nxB < HW && (unsigned)nyB < HW) ? Vc[cb + nxB * HW + nyB] : vbc;
            aA += w * vA;
            aB += w * vB;
        }
    }
    out[cb + pA] = aA * invZ;
    out[cb + pB] = aB * invZ;
}

extern "C" void kernel_launch(void* const* d_in, const int* in_sizes, int n_in,
                              void* d_out, int out_size, void* d_ws, size_t ws_size,
                              hipStream_t stream) {
    const float* x   = (const float*)d_in[0];
    const float* q_w = (const float*)d_in[1];
    const float* q_b = (const float*)d_in[2];
    const float* k_w = (const float*)d_in[3];
    const float* k_b = (const float*)d_in[4];
    const float* v_w = (const float*)d_in[5];
    const float* v_b = (const float*)d_in[6];
    const float* h_pos = (const float*)d_in[7];
    const float* w_pos = (const float*)d_in[8];
    float* out = (float*)d_out;
    float* ws  = (float*)d_ws;     // [Q | K | V | bias49] = 3*512*3136 + 49 floats

    float* bias49 = ws + (size_t)3 * C_DIM * P_DIM;

    // GEMM: 3 matrices * 4 o-blocks(128) * 49 p-tiles(64) = 588 blocks of 8 waves.
    qkv_gemm_wmma<<<588, 256, 0, stream>>>(x, q_w, q_b, k_w, k_b, v_w, v_b, ws);
    pos_bias_kernel<<<1, 64, 0, stream>>>(h_pos, w_pos, bias49);
    // Attention: 8*3136 = 25088 waves; 8 waves/block.
    local_attn_kernel<<<3136, 256, 0, stream>>>(ws, k_b, v_b, out);
}